// AnomalyNet_15118284881954
// MI455X (gfx1250) — compile-verified
//
#include <hip/hip_runtime.h>
#include <hip/hip_bf16.h>
#include <math.h>

// ---------------------------------------------------------------------------
// AnomalyNet forward for MI455X (gfx1250, wave32, WMMA).
//
// Pipeline:
//   1. h = tanh(x @ W_in + b_in)            -- WMMA f32 16x16x4 GEMM (fused)
//   2. Tx1 = A h ; Tx2 = A Tx1 ; Tx3 = A Tx2 (COO SpMM, atomic scatter-add,
//                                              prefetched edge stream)
//   3. low/high combos (in-place over Tx3/Tx2) + per-row L2 norm dn
//   4. hd = A dn ; global min/max of hd
//   5. blend + relu + [128x7] GEMM + log_softmax  -> d_out [N,7] f32
// ---------------------------------------------------------------------------

#define NN 50000
#define EE 800000
#define CIN 256
#define COUT 128
#define NCLS 7
#define XS_STRIDE 260   // 16-row x-tile LDS stride (260 % 64 == 4 -> conflict-free)

typedef float v2f __attribute__((ext_vector_type(2)));
typedef float v8f __attribute__((ext_vector_type(8)));

// ---------------------------------------------------------------------------
// Kernel 1: h = tanh(x @ W_in + b_in) with V_WMMA_F32_16X16X4_F32.
// Grid: 3125 blocks (one 16-row tile each), 256 threads = 8 waves.
// Wave w computes the 16x16 output tile at columns [16w, 16w+16).
// A-frag layout (32-bit A 16x4): lanes 0-15 hold M=lane, K={k0,k0+1};
//                                lanes 16-31 hold M=lane-16, K={k0+2,k0+3}.
// B-frag layout mirrors A (K selected by lane half, N = lane%16).
// C/D layout: vgpr r holds M = r + (lane>=16 ? 8 : 0), N = lane%16.
// ---------------------------------------------------------------------------
__global__ __launch_bounds__(256)
void gemm_tanh_wmma(const float* __restrict__ X, const float* __restrict__ W,
                    const float* __restrict__ bias, float* __restrict__ H) {
    __shared__ float xs[16 * XS_STRIDE];

    const int rowBase = blockIdx.x * 16;

    // Cooperatively stage the 16x256 f32 x-tile (16 KB) into LDS as float4.
    for (int i = threadIdx.x; i < 16 * (CIN / 4); i += 256) {
        const int r  = i >> 6;          // 64 float4 per row
        const int c4 = i & 63;
        const float4 v =
            reinterpret_cast<const float4*>(X + (size_t)(rowBase + r) * CIN)[c4];
        float* dst = &xs[r * XS_STRIDE + c4 * 4];
        dst[0] = v.x; dst[1] = v.y; dst[2] = v.z; dst[3] = v.w;
    }
    __syncthreads();

    const int wave = threadIdx.x >> 5;      // 0..7 -> N tile
    const int lane = threadIdx.x & 31;
    const int m    = lane & 15;
    const int koff = (lane >> 4) << 1;      // 0 for lanes 0-15, 2 for 16-31
    const int col  = wave * 16 + m;

    const float* __restrict__ wcol = W + col;   // W[k][col] = W[k*COUT + col]
    v8f acc = {};

    #pragma unroll 4
    for (int k0 = 0; k0 < CIN; k0 += 4) {
        v2f a = *reinterpret_cast<const v2f*>(&xs[m * XS_STRIDE + k0 + koff]);
        v2f b;
        b.x = wcol[(size_t)(k0 + koff)     * COUT];
        b.y = wcol[(size_t)(k0 + koff + 1) * COUT];
        // D = A(16x4,f32) * B(4x16,f32) + C(16x16,f32)
        acc = __builtin_amdgcn_wmma_f32_16x16x4_f32(
            /*neg_a=*/false, a, /*neg_b=*/false, b,
            /*c_mod=*/(short)0, acc, /*reuse_a=*/false, /*reuse_b=*/false);
    }

    const float bv    = bias[col];
    const int   mBase = rowBase + ((lane >> 4) << 3);
    #pragma unroll
    for (int r = 0; r < 8; ++r) {
        H[(size_t)(mBase + r) * COUT + col] = tanhf(acc[r] + bv);
    }
}

// ---------------------------------------------------------------------------
// Zero-fill (workspace is poisoned with 0xAA by the harness).
// ---------------------------------------------------------------------------
__global__ void zero_f32(float* __restrict__ p, long n) {
    long i      = (long)blockIdx.x * blockDim.x + threadIdx.x;
    long stride = (long)gridDim.x * blockDim.x;
    for (; i < n; i += stride) p[i] = 0.0f;
}

// ---------------------------------------------------------------------------
// COO SpMM over 128 channels: out[rows[e]] += vals[e] * src[cols[e]].
// One wave per edge; lane owns channels [4*lane, 4*lane+4).
// 16B gather + 4 global f32 atomics (RMW stays in L2; the 25.6 MB
// accumulation buffer fits in the 192 MB L2). Next iteration's edge data is
// prefetched (global_prefetch_b8) to hide index-fetch latency behind the
// gather/atomic phase.
// ---------------------------------------------------------------------------
__global__ __launch_bounds__(256)
void spmm128(const int* __restrict__ rows, const int* __restrict__ cols,
             const float* __restrict__ vals, const float* __restrict__ src,
             float* __restrict__ out, int numE) {
    const int lane   = threadIdx.x & 31;
    const int wid    = (blockIdx.x * (blockDim.x >> 5)) + (threadIdx.x >> 5);
    const int nwaves = gridDim.x * (blockDim.x >> 5);
    for (int e = wid; e < numE; e += nwaves) {
        const int en = e + nwaves;
        if (en < numE) {
            __builtin_prefetch(&rows[en], 0, 1);
            __builtin_prefetch(&cols[en], 0, 1);
            __builtin_prefetch(&vals[en], 0, 1);
        }
        const int   r = rows[e];
        const int   c = cols[e];
        const float v = vals[e];
        const float4 s = reinterpret_cast<const float4*>(src + (size_t)c * COUT)[lane];
        float* o = out + (size_t)r * COUT + lane * 4;
        atomicAdd(o + 0, v * s.x);
        atomicAdd(o + 1, v * s.y);
        atomicAdd(o + 2, v * s.z);
        atomicAdd(o + 3, v * s.w);
    }
}

// ---------------------------------------------------------------------------
// Fused: low/high polynomial combos (in place: low -> t3, high -> t2) and
// dn[row] = || low[row] - h[row] ||_2. One wave per row, wave32 shfl reduce.
// ---------------------------------------------------------------------------
__global__ __launch_bounds__(256)
void lowhigh_dn(const float* __restrict__ h,  const float* __restrict__ t1,
                float* __restrict__ t2,       float* __restrict__ t3,
                const float* __restrict__ dptr, const float* __restrict__ aptr,
                float* __restrict__ dn) {
    const int lane = threadIdx.x & 31;
    const int row  = blockIdx.x * (blockDim.x >> 5) + (threadIdx.x >> 5); // < NN exactly

    const float d  = dptr[0];
    const float av = aptr[0];
    const float cl2 = -3.0f * d - av;
    const float cl1 = 3.0f * d * d + 2.0f * d * av;
    const float cl0 = -(d * d * d + d * d * av);
    const float ch2 = -3.0f * d + av;
    const float ch1 = 3.0f * d * d - 2.0f * d * av;
    const float ch0 = d * d * av - d * d * d;

    const size_t base = (size_t)row * COUT + lane * 4;
    const float4 H  = *reinterpret_cast<const float4*>(h  + base);
    const float4 T1 = *reinterpret_cast<const float4*>(t1 + base);
    const float4 T2 = *reinterpret_cast<const float4*>(t2 + base);
    const float4 T3 = *reinterpret_cast<const float4*>(t3 + base);

    float4 lo, hi;
    lo.x = T3.x + cl2 * T2.x + cl1 * T1.x + cl0 * H.x;
    lo.y = T3.y + cl2 * T2.y + cl1 * T1.y + cl0 * H.y;
    lo.z = T3.z + cl2 * T2.z + cl1 * T1.z + cl0 * H.z;
    lo.w = T3.w + cl2 * T2.w + cl1 * T1.w + cl0 * H.w;
    hi.x = T3.x + ch2 * T2.x + ch1 * T1.x + ch0 * H.x;
    hi.y = T3.y + ch2 * T2.y + ch1 * T1.y + ch0 * H.y;
    hi.z = T3.z + ch2 * T2.z + ch1 * T1.z + ch0 * H.z;
    hi.w = T3.w + ch2 * T2.w + ch1 * T1.w + ch0 * H.w;

    *reinterpret_cast<float4*>(t3 + base) = lo;   // low
    *reinterpret_cast<float4*>(t2 + base) = hi;   // high

    const float dx = lo.x - H.x, dy = lo.y - H.y, dz = lo.z - H.z, dw = lo.w - H.w;
    float ss = dx * dx + dy * dy + dz * dz + dw * dw;
    #pragma unroll
    for (int off = 16; off > 0; off >>= 1) ss += __shfl_xor(ss, off, 32);
    if (lane == 0) dn[row] = sqrtf(ss);
}

// ---------------------------------------------------------------------------
// Scalar COO SpMV: hd[rows[e]] += vals[e] * dn[cols[e]].
// ---------------------------------------------------------------------------
__global__ void spmm1(const int* __restrict__ rows, const int* __restrict__ cols,
                      const float* __restrict__ vals, const float* __restrict__ dn,
                      float* __restrict__ hd, int numE) {
    int i      = blockIdx.x * blockDim.x + threadIdx.x;
    int stride = gridDim.x * blockDim.x;
    for (; i < numE; i += stride)
        atomicAdd(&hd[rows[i]], vals[i] * dn[cols[i]]);
}

// ---------------------------------------------------------------------------
// Global min/max of hd: two-stage deterministic reduction.
// ---------------------------------------------------------------------------
__global__ __launch_bounds__(256)
void minmax_partial(const float* __restrict__ hd, int n,
                    float* __restrict__ pmin, float* __restrict__ pmax) {
    __shared__ float smin[256], smax[256];
    float mn = 3.4e38f, mx = -3.4e38f;
    for (int i = blockIdx.x * blockDim.x + threadIdx.x; i < n;
         i += gridDim.x * blockDim.x) {
        const float v = hd[i];
        mn = fminf(mn, v);
        mx = fmaxf(mx, v);
    }
    smin[threadIdx.x] = mn; smax[threadIdx.x] = mx;
    __syncthreads();
    for (int s = 128; s > 0; s >>= 1) {
        if (threadIdx.x < s) {
            smin[threadIdx.x] = fminf(smin[threadIdx.x], smin[threadIdx.x + s]);
            smax[threadIdx.x] = fmaxf(smax[threadIdx.x], smax[threadIdx.x + s]);
        }
        __syncthreads();
    }
    if (threadIdx.x == 0) { pmin[blockIdx.x] = smin[0]; pmax[blockIdx.x] = smax[0]; }
}

__global__ __launch_bounds__(256)
void minmax_final(const float* __restrict__ pmin, const float* __restrict__ pmax,
                  float* __restrict__ mm) {
    __shared__ float smin[256], smax[256];
    smin[threadIdx.x] = pmin[threadIdx.x];
    smax[threadIdx.x] = pmax[threadIdx.x];
    __syncthreads();
    for (int s = 128; s > 0; s >>= 1) {
        if (threadIdx.x < s) {
            smin[threadIdx.x] = fminf(smin[threadIdx.x], smin[threadIdx.x + s]);
            smax[threadIdx.x] = fmaxf(smax[threadIdx.x], smax[threadIdx.x + s]);
        }
        __syncthreads();
    }
    if (threadIdx.x == 0) { mm[0] = smin[0]; mm[1] = smax[0]; }
}

// ---------------------------------------------------------------------------
// Fused output: blend + relu + [128x7] GEMV + log_softmax. One wave per row;
// W_out (128x7) + b_out staged in LDS per block.
// ---------------------------------------------------------------------------
__global__ __launch_bounds__(256)
void final_out(const float* __restrict__ low, const float* __restrict__ high,
               const float* __restrict__ hd,  const float* __restrict__ mm,
               const float* __restrict__ Wout, const float* __restrict__ bout,
               float* __restrict__ out) {
    __shared__ float wsh[COUT * NCLS];
    __shared__ float bsh[NCLS];
    for (int i = threadIdx.x; i < COUT * NCLS; i += 256) wsh[i] = Wout[i];
    if (threadIdx.x < NCLS) bsh[threadIdx.x] = bout[threadIdx.x];
    __syncthreads();

    const int lane = threadIdx.x & 31;
    const int row  = blockIdx.x * (blockDim.x >> 5) + (threadIdx.x >> 5); // < NN exactly

    const float mn  = mm[0];
    const float mx  = mm[1];
    const float nrm = (hd[row] - mn) / (mx - mn);
    const float wlo = 1.0f - nrm;

    const size_t base = (size_t)row * COUT + lane * 4;
    const float4 L = *reinterpret_cast<const float4*>(low  + base);
    const float4 H = *reinterpret_cast<const float4*>(high + base);
    float f[4];
    f[0] = fmaxf(wlo * L.x + nrm * H.x, 0.0f);
    f[1] = fmaxf(wlo * L.y + nrm * H.y, 0.0f);
    f[2] = fmaxf(wlo * L.z + nrm * H.z, 0.0f);
    f[3] = fmaxf(wlo * L.w + nrm * H.w, 0.0f);

    float acc[NCLS];
    #pragma unroll
    for (int j = 0; j < NCLS; ++j) acc[j] = 0.0f;
    #pragma unroll
    for (int q = 0; q < 4; ++q) {
        const int k = lane * 4 + q;
        #pragma unroll
        for (int j = 0; j < NCLS; ++j) acc[j] += f[q] * wsh[k * NCLS + j];
    }
    #pragma unroll
    for (int off = 16; off > 0; off >>= 1) {
        #pragma unroll
        for (int j = 0; j < NCLS; ++j) acc[j] += __shfl_xor(acc[j], off, 32);
    }

    if (lane == 0) {
        float y[NCLS];
        float m = -3.4e38f;
        #pragma unroll
        for (int j = 0; j < NCLS; ++j) { y[j] = acc[j] + bsh[j]; m = fmaxf(m, y[j]); }
        float s = 0.0f;
        #pragma unroll
        for (int j = 0; j < NCLS; ++j) s += __expf(y[j] - m);
        const float lse = __logf(s);
        #pragma unroll
        for (int j = 0; j < NCLS; ++j) out[(size_t)row * NCLS + j] = y[j] - m - lse;
    }
}

// ---------------------------------------------------------------------------
// Host-side orchestration (graph-capture safe: kernels only, all on stream).
// Workspace layout (floats): h | t1 | t2 | t3 | dn | hd | pmin | pmax | mm
// Total ~103 MB.
// ---------------------------------------------------------------------------
extern "C" void kernel_launch(void* const* d_in, const int* in_sizes, int n_in,
                              void* d_out, int out_size, void* d_ws, size_t ws_size,
                              hipStream_t stream) {
    const float* x     = (const float*)d_in[0];
    const float* vals  = (const float*)d_in[1];
    const float* W_in  = (const float*)d_in[2];
    const float* b_in  = (const float*)d_in[3];
    const float* delta = (const float*)d_in[4];
    const float* a     = (const float*)d_in[5];
    const float* W_out = (const float*)d_in[6];
    const float* b_out = (const float*)d_in[7];
    const int*   rows  = (const int*)d_in[8];
    const int*   cols  = (const int*)d_in[9];
    float*       out   = (float*)d_out;

    const long NB = (long)NN * COUT;          // 6.4M floats per tile buffer
    float* h    = (float*)d_ws;
    float* t1   = h  + NB;
    float* t2   = t1 + NB;
    float* t3   = t2 + NB;
    float* dn   = t3 + NB;
    float* hd   = dn + NN;
    float* pmin = hd + NN;
    float* pmax = pmin + 256;
    float* mm   = pmax + 256;

    // 1. h = tanh(x @ W_in + b_in)   [WMMA f32 16x16x4]
    gemm_tanh_wmma<<<NN / 16, 256, 0, stream>>>(x, W_in, b_in, h);

    // 2. Tx1, Tx2, Tx3 chained COO SpMMs
    zero_f32<<<512, 256, 0, stream>>>(t1, NB);
    spmm128<<<2048, 256, 0, stream>>>(rows, cols, vals, h, t1, EE);
    zero_f32<<<512, 256, 0, stream>>>(t2, NB);
    spmm128<<<2048, 256, 0, stream>>>(rows, cols, vals, t1, t2, EE);
    zero_f32<<<512, 256, 0, stream>>>(t3, NB);
    spmm128<<<2048, 256, 0, stream>>>(rows, cols, vals, t2, t3, EE);

    // 3. low -> t3, high -> t2, dn
    lowhigh_dn<<<NN / 8, 256, 0, stream>>>(h, t1, t2, t3, delta, a, dn);

    // 4. hd = A dn, then global min/max
    zero_f32<<<64, 256, 0, stream>>>(hd, NN);
    spmm1<<<1024, 256, 0, stream>>>(rows, cols, vals, dn, hd, EE);
    minmax_partial<<<256, 256, 0, stream>>>(hd, NN, pmin, pmax);
    minmax_final<<<1, 256, 0, stream>>>(pmin, pmax, mm);

    // 5. blend + relu + GEMV + log_softmax
    final_out<<<NN / 8, 256, 0, stream>>>(t3, t2, hd, mm, W_out, b_out, out);
}